// SubgraphSelector_19490561589475
// MI455X (gfx1250) — compile-verified
//
#include <hip/hip_runtime.h>
#include <math.h>

#define CH 64   // IN_CH == HID == 64

typedef __attribute__((ext_vector_type(2))) float v2f;
typedef __attribute__((ext_vector_type(8))) float v8f;

// ---------------------------------------------------------------------------
// Degree / normalization kernels
// ---------------------------------------------------------------------------
__global__ void deg_init_kernel(float* __restrict__ deg, int n) {
    int i = blockIdx.x * blockDim.x + threadIdx.x;
    if (i < n) deg[i] = 1.0f;                 // self-loop contributes 1
}

__global__ void deg_count_kernel(const int* __restrict__ dst,
                                 float* __restrict__ deg, int e) {
    int i = blockIdx.x * blockDim.x + threadIdx.x;
    if (i < e) atomicAdd(&deg[dst[i]], 1.0f);
}

__global__ void deg_rsqrt_kernel(float* __restrict__ deg, int n) {
    int i = blockIdx.x * blockDim.x + threadIdx.x;
    if (i < n) deg[i] = rsqrtf(deg[i]);       // in-place -> dinv
}

// ---------------------------------------------------------------------------
// WMMA GEMM:  h = X @ W  (X: [N,64] f32, W: [64,64] f32)
// Epilogue:   g = h * dinv[row]   ->  G  (messages)  and  Acc (self-loop seed)
// One 256-thread block = 8 wave32; each wave computes a 16x64 output tile
// using V_WMMA_F32_16X16X4_F32 (exact fp32 path).
// ---------------------------------------------------------------------------
__global__ void gcn_gemm_wmma(const float* __restrict__ X,
                              const float* __restrict__ W,
                              const float* __restrict__ dinv,
                              float* __restrict__ G,
                              float* __restrict__ Acc,
                              int nRowTiles) {
    __shared__ float Wt[CH][CH + 2];          // transposed W, padded (bank safety)

    const int tid = threadIdx.x;
    for (int i = tid; i < CH * CH; i += 256) {
        int k = i >> 6, c = i & 63;
        Wt[c][k] = W[i];                      // Wt[c][k] = W[k][c]
    }
    __syncthreads();

    const int wave    = tid >> 5;
    const int lane    = tid & 31;
    const int rowTile = blockIdx.x * 8 + wave;      // wave-uniform
    if (rowTile >= nRowTiles) return;               // no intra-wave divergence

    const int row0  = rowTile << 4;
    const int r     = lane & 15;                    // A-row / B-col within tile
    const int khalf = (lane >> 4) << 1;             // 0 or 2 (K sub-pair)

    const float* xrow = X + (size_t)(row0 + r) * CH;

    v8f acc0 = {}, acc1 = {}, acc2 = {}, acc3 = {};

    #pragma unroll 4
    for (int kk = 0; kk < 16; ++kk) {
        const int k = (kk << 2) + khalf;
        // A fragment: lane holds {X[row][k], X[row][k+1]}
        v2f a = *(const v2f*)(xrow + k);
        // B fragments: lane holds {W[k][c], W[k+1][c]} = contiguous in Wt[c]
        v2f b0 = *(const v2f*)(&Wt[r +  0][k]);
        v2f b1 = *(const v2f*)(&Wt[r + 16][k]);
        v2f b2 = *(const v2f*)(&Wt[r + 32][k]);
        v2f b3 = *(const v2f*)(&Wt[r + 48][k]);
        acc0 = __builtin_amdgcn_wmma_f32_16x16x4_f32(false, a, false, b0, (short)0, acc0, false, false);
        acc1 = __builtin_amdgcn_wmma_f32_16x16x4_f32(false, a, false, b1, (short)0, acc1, false, false);
        acc2 = __builtin_amdgcn_wmma_f32_16x16x4_f32(false, a, false, b2, (short)0, acc2, false, false);
        acc3 = __builtin_amdgcn_wmma_f32_16x16x4_f32(false, a, false, b3, (short)0, acc3, false, false);
    }

    // D layout: lane<16 -> VGPR i = D[M=i][N=lane]; lane>=16 -> D[M=8+i][N=lane-16]
    const int mbase = (lane >> 4) << 3;             // 0 or 8
    #pragma unroll
    for (int i = 0; i < 8; ++i) {
        const int row = row0 + mbase + i;
        const float dv = dinv[row];
        float* grow = G   + (size_t)row * CH;
        float* arow = Acc + (size_t)row * CH;
        float g0 = acc0[i] * dv;
        float g1 = acc1[i] * dv;
        float g2 = acc2[i] * dv;
        float g3 = acc3[i] * dv;
        grow[r +  0] = g0;  arow[r +  0] = g0;
        grow[r + 16] = g1;  arow[r + 16] = g1;
        grow[r + 32] = g2;  arow[r + 32] = g2;
        grow[r + 48] = g3;  arow[r + 48] = g3;
    }
}

// ---------------------------------------------------------------------------
// Edge scatter-add:  Acc[dst] += G[src]   (16 lanes per edge, float4 each)
// ---------------------------------------------------------------------------
__global__ void edge_agg_kernel(const int* __restrict__ src,
                                const int* __restrict__ dst,
                                const float* __restrict__ G,
                                float* __restrict__ Acc, int e) {
    long long t = (long long)blockIdx.x * blockDim.x + threadIdx.x;
    int edge = (int)(t >> 4);
    int part = (int)(t & 15);
    if (edge >= e) return;
    int s = src[edge];
    int d = dst[edge];
    const float4 v = *(const float4*)(G + (size_t)s * CH + part * 4);
    float* ap = Acc + (size_t)d * CH + part * 4;
    atomicAdd(ap + 0, v.x);
    atomicAdd(ap + 1, v.y);
    atomicAdd(ap + 2, v.z);
    atomicAdd(ap + 3, v.w);
}

// ---------------------------------------------------------------------------
// Post: out = relu(Acc * dinv[node] + bias[channel])
// ---------------------------------------------------------------------------
__global__ void post_bias_relu_kernel(const float* __restrict__ Acc,
                                      const float* __restrict__ dinv,
                                      const float* __restrict__ bias,
                                      float* __restrict__ out, int n64) {
    int i = blockIdx.x * blockDim.x + threadIdx.x;
    if (i < n64) {
        int node = i >> 6, c = i & 63;
        float v = Acc[i] * dinv[node] + bias[c];
        out[i] = v > 0.0f ? v : 0.0f;
    }
}

// ---------------------------------------------------------------------------
// FC head:  p = sigmoid(H @ Wfc + bfc)    (one thread per node, 64-dot)
// ---------------------------------------------------------------------------
__global__ void fc_sigmoid_kernel(const float* __restrict__ H,
                                  const float* __restrict__ Wfc,
                                  const float* __restrict__ bfc,
                                  float* __restrict__ out, int n) {
    __shared__ float w[CH];
    if (threadIdx.x < CH) w[threadIdx.x] = Wfc[threadIdx.x];
    __syncthreads();
    int i = blockIdx.x * blockDim.x + threadIdx.x;
    if (i >= n) return;
    const float* h = H + (size_t)i * CH;
    float s = bfc[0];
    #pragma unroll
    for (int j = 0; j < CH; j += 4) {
        float4 v = *(const float4*)(h + j);
        s += v.x * w[j] + v.y * w[j + 1] + v.z * w[j + 2] + v.w * w[j + 3];
    }
    out[i] = 1.0f / (1.0f + __expf(-s));
}

// ---------------------------------------------------------------------------
// Launcher
// ---------------------------------------------------------------------------
extern "C" void kernel_launch(void* const* d_in, const int* in_sizes, int n_in,
                              void* d_out, int out_size, void* d_ws, size_t ws_size,
                              hipStream_t stream) {
    const float* x   = (const float*)d_in[0];
    const int*   ei  = (const int*)d_in[1];
    const float* W1  = (const float*)d_in[2];
    const float* b1  = (const float*)d_in[3];
    const float* W2  = (const float*)d_in[4];
    const float* b2  = (const float*)d_in[5];
    const float* Wfc = (const float*)d_in[6];
    const float* bfc = (const float*)d_in[7];
    float* out = (float*)d_out;

    const int N = in_sizes[0] / CH;     // 100000
    const int E = in_sizes[1] / 2;      // 1600000
    const int* src = ei;
    const int* dst = ei + E;

    // workspace carve-up (bytes): dinv | G | Acc | H
    char* ws = (char*)d_ws;
    float* dinv = (float*)ws;  ws += ((size_t)N * 4 + 255) & ~(size_t)255;
    float* G    = (float*)ws;  ws += (size_t)N * CH * 4;
    float* Acc  = (float*)ws;  ws += (size_t)N * CH * 4;
    float* H    = (float*)ws;

    const int n64 = N * CH;

    // --- symmetric normalization: dinv = rsqrt(in_degree + 1) ---
    deg_init_kernel<<<(N + 255) / 256, 256, 0, stream>>>(dinv, N);
    deg_count_kernel<<<(E + 255) / 256, 256, 0, stream>>>(dst, dinv, E);
    deg_rsqrt_kernel<<<(N + 255) / 256, 256, 0, stream>>>(dinv, N);

    const int nRowTiles  = N / 16;                 // N divisible by 16
    const int gemmBlocks = (nRowTiles + 7) / 8;
    const long long eThreads = (long long)E * 16;
    const int edgeBlocks = (int)((eThreads + 255) / 256);

    // --- layer 1 ---
    gcn_gemm_wmma<<<gemmBlocks, 256, 0, stream>>>(x, W1, dinv, G, Acc, nRowTiles);
    edge_agg_kernel<<<edgeBlocks, 256, 0, stream>>>(src, dst, G, Acc, E);
    post_bias_relu_kernel<<<(n64 + 255) / 256, 256, 0, stream>>>(Acc, dinv, b1, H, n64);

    // --- layer 2 ---
    gcn_gemm_wmma<<<gemmBlocks, 256, 0, stream>>>(H, W2, dinv, G, Acc, nRowTiles);
    edge_agg_kernel<<<edgeBlocks, 256, 0, stream>>>(src, dst, G, Acc, E);
    post_bias_relu_kernel<<<(n64 + 255) / 256, 256, 0, stream>>>(Acc, dinv, b2, H, n64);

    // --- FC + sigmoid ---
    fc_sigmoid_kernel<<<(N + 255) / 256, 256, 0, stream>>>(H, Wfc, bfc, out, N);
}